// MyGCNEdge_50242527429293
// MI455X (gfx1250) — compile-verified
//
#include <hip/hip_runtime.h>

// ---------------------------------------------------------------------------
// GCNConv(aggr='max') x2 + linear head for MI455X (gfx1250, wave32).
// Dominant cost: edge scatter-max (L2-resident atomics). Dense transforms use
// V_WMMA_F32_16X16X4_F32 (full f32 precision, matches reference numerics).
// ---------------------------------------------------------------------------

typedef __attribute__((ext_vector_type(2))) float v2f;
typedef __attribute__((ext_vector_type(8))) float v8f;

#define FEAT 240
#define FOUT 80
#define KEY_NEG_INF 0x007FFFFFu  // ordered-key encoding of -inf

// Monotone float <-> u32 key mapping (order-preserving, so u32 atomicMax
// == float max; lowers to hardware GLOBAL_ATOMIC_MAX_U32, no CAS loop).
__device__ __forceinline__ unsigned f2key(float f) {
    unsigned u = __float_as_uint(f);
    return (u & 0x80000000u) ? ~u : (u | 0x80000000u);
}
__device__ __forceinline__ float key2f(unsigned k) {
    unsigned u = (k & 0x80000000u) ? (k ^ 0x80000000u) : ~k;
    return __uint_as_float(u);
}

// ---------------------------------------------------------------------------
__global__ void fill_u32_kernel(unsigned* __restrict__ p, unsigned v, int n) {
    int i = blockIdx.x * blockDim.x + threadIdx.x;
    if (i < n) p[i] = v;
}

// deg[col] += ew  (edges), deg[n] += 1 (self loops). deg pre-zeroed.
__global__ void degree_kernel(const long long* __restrict__ ei,
                              const float* __restrict__ ew,
                              float* __restrict__ deg, int E, int N) {
    int i = blockIdx.x * blockDim.x + threadIdx.x;
    if (i >= E + N) return;
    if (i < E) {
        long long c = ei[(size_t)E + i];
        atomicAdd(&deg[c], ew[i]);
    } else {
        atomicAdd(&deg[i - E], 1.0f);
    }
}

// deg -> dis = deg>0 ? rsqrt(deg) : 0 (in place)
__global__ void dis_kernel(float* __restrict__ deg, int N) {
    int i = blockIdx.x * blockDim.x + threadIdx.x;
    if (i >= N) return;
    float d = deg[i];
    deg[i] = (d > 0.0f) ? rsqrtf(d) : 0.0f;
}

// norm[e] = dis[row]*ew*dis[col]; self loops: dis[n]^2. Shared by both layers.
__global__ void norm_kernel(const long long* __restrict__ ei,
                            const float* __restrict__ ew,
                            const float* __restrict__ dis,
                            float* __restrict__ nrm, int E, int N) {
    int i = blockIdx.x * blockDim.x + threadIdx.x;
    if (i >= E + N) return;
    if (i < E) {
        long long r = ei[i];
        long long c = ei[(size_t)E + i];
        nrm[i] = dis[r] * ew[i] * dis[c];
    } else {
        float d = dis[i - E];
        nrm[i] = d * d;
    }
}

// h1 = x @ W1   (K = 5, too small for WMMA; 0.12 GFLOP total)
__global__ void linear1_kernel(const float* __restrict__ x,
                               const float* __restrict__ W,
                               float* __restrict__ out, int N) {
    int i = blockIdx.x * blockDim.x + threadIdx.x;
    if (i >= N * FEAT) return;
    int n = i / FEAT;
    int j = i - n * FEAT;
    const float* xr = x + (size_t)n * 5;
    float s = 0.0f;
#pragma unroll
    for (int k = 0; k < 5; ++k) s += xr[k] * W[k * FEAT + j];
    out[i] = s;
}

// One wave per edge: lanes stride the 240 features (coalesced row gather),
// atomicMax with ordered u32 keys into agg[col].
__global__ void scatter_max_kernel(const float* __restrict__ h,
                                   const float* __restrict__ nrm,
                                   const long long* __restrict__ ei,
                                   unsigned* __restrict__ agg, int E, int N) {
    int wave = (blockIdx.x * blockDim.x + threadIdx.x) >> 5;
    int lane = threadIdx.x & 31;
    if (wave >= E + N) return;
    long long r, c;
    if (wave < E) {
        r = ei[wave];
        c = ei[(size_t)E + wave];
    } else {
        r = c = wave - E;
    }
    float w = nrm[wave];
    const float* hr = h + (size_t)r * FEAT;
    unsigned* ag = agg + (size_t)c * FEAT;
#pragma unroll
    for (int f = lane; f < FEAT; f += 32) {
        unsigned key = f2key(w * hr[f]);
        atomicMax(ag + f, key);
    }
}

// decode ordered keys, add bias, ReLU; store float bits in place.
__global__ void bias_relu_kernel(unsigned* __restrict__ buf,
                                 const float* __restrict__ bias, int total) {
    int i = blockIdx.x * blockDim.x + threadIdx.x;
    if (i >= total) return;
    float v = key2f(buf[i]) + bias[i % FEAT];
    v = v > 0.0f ? v : 0.0f;
    buf[i] = __float_as_uint(v);
}

// ---------------------------------------------------------------------------
// f32 WMMA GEMM: C[M x NC] = A[M x K] @ B[K x NC] (+ bias). One wave per
// 16x16 output tile; K-loop in steps of 4 via V_WMMA_F32_16X16X4_F32.
// A 16x4 fragment layout (ISA 7.12.2): lanes 0-15 -> K={0,1}, 16-31 -> K={2,3}.
// C/D: VGPR r holds row r (lanes 0-15) / row r+8 (lanes 16-31).
// M, NC, K here are all multiples of 16 -> EXEC all-ones, no tails.
// ---------------------------------------------------------------------------
template <int K, int NC, bool BIAS>
__global__ void gemm_wmma_kernel(const float* __restrict__ A,
                                 const float* __restrict__ B,
                                 const float* __restrict__ bias,
                                 float* __restrict__ C, int M) {
    const int wave = (blockIdx.x * blockDim.x + threadIdx.x) >> 5;
    const int lane = threadIdx.x & 31;
    constexpr int NT = NC / 16;
    const int mt = wave / NT;
    const int nt = wave - mt * NT;
    if (mt * 16 >= M) return;  // wave-uniform

    const int half = lane >> 4;  // 0: K pair {0,1}, 1: K pair {2,3}
    const int l16 = lane & 15;
    const int koff = half * 2;

    const float* arow = A + (size_t)(mt * 16 + l16) * K;
    const float* bcol = B + nt * 16 + l16;

    v8f acc = {};
#pragma unroll 4
    for (int k = 0; k < K; k += 4) {
        v2f a, b;
        a.x = arow[k + koff];
        a.y = arow[k + koff + 1];
        b.x = bcol[(size_t)(k + koff) * NC];
        b.y = bcol[(size_t)(k + koff + 1) * NC];
        acc = __builtin_amdgcn_wmma_f32_16x16x4_f32(
            /*neg_a=*/false, a, /*neg_b=*/false, b,
            /*c_mod=*/(short)0, acc, /*reuse_a=*/false, /*reuse_b=*/false);
    }

    const float bv = BIAS ? bias[nt * 16 + l16] : 0.0f;
    float* crow = C + (size_t)(mt * 16 + half * 8) * NC + nt * 16 + l16;
#pragma unroll
    for (int r = 0; r < 8; ++r) crow[(size_t)r * NC] = acc[r] + bv;
}

// ---------------------------------------------------------------------------
static inline int nblk(long long threads) { return (int)((threads + 255) / 256); }

extern "C" void kernel_launch(void* const* d_in, const int* in_sizes, int n_in,
                              void* d_out, int out_size, void* d_ws, size_t ws_size,
                              hipStream_t stream) {
    const float*     x  = (const float*)d_in[0];
    const long long* ei = (const long long*)d_in[1];  // int64 [2, E]
    const float*     ew = (const float*)d_in[2];
    const float*     W1 = (const float*)d_in[3];
    const float*     b1 = (const float*)d_in[4];
    const float*     W2 = (const float*)d_in[5];
    const float*     b2 = (const float*)d_in[6];
    const float*     We = (const float*)d_in[7];
    const float*     be = (const float*)d_in[8];
    float*           out = (float*)d_out;

    const int N = in_sizes[0] / 5;
    const int E = in_sizes[2];

    // Workspace: deg[N] | norm[E+N] | hA[N*240] | aggB[N*240]  (~103 MB)
    float*    deg  = (float*)d_ws;
    float*    nrm  = deg + N;
    float*    hA   = nrm + ((size_t)E + N);
    unsigned* aggB = (unsigned*)(hA + (size_t)N * FEAT);

    const int NF = N * FEAT;

    // --- shared normalization ------------------------------------------------
    fill_u32_kernel<<<nblk(N), 256, 0, stream>>>((unsigned*)deg, 0u, N);
    degree_kernel<<<nblk((long long)E + N), 256, 0, stream>>>(ei, ew, deg, E, N);
    dis_kernel<<<nblk(N), 256, 0, stream>>>(deg, N);
    norm_kernel<<<nblk((long long)E + N), 256, 0, stream>>>(ei, ew, deg, nrm, E, N);

    // --- layer 1: linear (K=5), scatter-max, bias+ReLU ----------------------
    linear1_kernel<<<nblk(NF), 256, 0, stream>>>(x, W1, hA, N);
    fill_u32_kernel<<<nblk(NF), 256, 0, stream>>>(aggB, KEY_NEG_INF, NF);
    scatter_max_kernel<<<nblk(((long long)E + N) * 32), 256, 0, stream>>>(hA, nrm, ei, aggB, E, N);
    bias_relu_kernel<<<nblk(NF), 256, 0, stream>>>(aggB, b1, NF);

    // --- layer 2: WMMA GEMM (240x240), scatter-max, bias+ReLU ---------------
    {
        long long waves = (long long)((N + 15) / 16) * (FEAT / 16);
        gemm_wmma_kernel<FEAT, FEAT, false>
            <<<nblk(waves * 32), 256, 0, stream>>>((const float*)aggB, W2, nullptr, hA, N);
    }
    fill_u32_kernel<<<nblk(NF), 256, 0, stream>>>(aggB, KEY_NEG_INF, NF);
    scatter_max_kernel<<<nblk(((long long)E + N) * 32), 256, 0, stream>>>(hA, nrm, ei, aggB, E, N);
    bias_relu_kernel<<<nblk(NF), 256, 0, stream>>>(aggB, b2, NF);

    // --- head: WMMA GEMM (240x80) + bias -> d_out ---------------------------
    {
        long long waves = (long long)((N + 15) / 16) * (FOUT / 16);
        gemm_wmma_kernel<FEAT, FOUT, true>
            <<<nblk(waves * 32), 256, 0, stream>>>((const float*)aggB, We, be, out, N);
    }
}